// BoundaryPredictor2_68427418960036
// MI455X (gfx1250) — compile-verified
//
#include <hip/hip_runtime.h>
#include <cstdint>

#define B_ 16
#define S_ 8192
#define D_ 512
#define SCALE_ 0.125f      // HEAD_DIM=64 -> 64^-0.5
#define PRIOR_ 0.2f

typedef float v2f __attribute__((ext_vector_type(2)));
typedef float v8f __attribute__((ext_vector_type(8)));

__device__ __forceinline__ float wave_sum(float v) {
#pragma unroll
  for (int off = 16; off >= 1; off >>= 1) v += __shfl_xor(v, off, 32);
  return v;
}

// ---------------- zero hsum scratch (ws is poisoned once, not re-poisoned) ----
__global__ void k_zero(float* __restrict__ hsum) {
  int i = blockIdx.x * blockDim.x + threadIdx.x;
  if (i < B_ * D_) hsum[i] = 0.f;
}

// ---------------- pass 1: per-token inverse L2 norm + per-batch hsum ---------
// grid (S/64, B), block 256 = 8 waves; each wave handles 8 tokens.
__global__ void k_norm_hsum(const float* __restrict__ hidden,
                            float* __restrict__ invnorm,
                            float* __restrict__ hsum) {
  __shared__ float lsum[D_];
  const int tid = threadIdx.x, lane = tid & 31, wave = tid >> 5;
  for (int i = tid; i < D_; i += 256) lsum[i] = 0.f;
  __syncthreads();
  const int b = blockIdx.y;
  const int s0 = blockIdx.x * 64 + wave * 8;
  float4 a0 = {0,0,0,0}, a1 = {0,0,0,0}, a2 = {0,0,0,0}, a3 = {0,0,0,0};
  for (int t = 0; t < 8; ++t) {
    const int s = s0 + t;
    const float* hp = hidden + ((size_t)b * S_ + s) * D_;
    const float4* hp4 = (const float4*)hp;
    float4 v0 = hp4[0 * 32 + lane];
    float4 v1 = hp4[1 * 32 + lane];
    float4 v2 = hp4[2 * 32 + lane];
    float4 v3 = hp4[3 * 32 + lane];
    if (t + 1 < 8) __builtin_prefetch(hp + D_ + lane * 16, 0, 1);
    float ss = v0.x*v0.x + v0.y*v0.y + v0.z*v0.z + v0.w*v0.w
             + v1.x*v1.x + v1.y*v1.y + v1.z*v1.z + v1.w*v1.w
             + v2.x*v2.x + v2.y*v2.y + v2.z*v2.z + v2.w*v2.w
             + v3.x*v3.x + v3.y*v3.y + v3.z*v3.z + v3.w*v3.w;
    ss = wave_sum(ss);
    const float rinv = 1.f / fmaxf(sqrtf(ss), 1e-12f);
    if (lane == 0) invnorm[b * S_ + s] = rinv;
    a0.x += v0.x * rinv; a0.y += v0.y * rinv; a0.z += v0.z * rinv; a0.w += v0.w * rinv;
    a1.x += v1.x * rinv; a1.y += v1.y * rinv; a1.z += v1.z * rinv; a1.w += v1.w * rinv;
    a2.x += v2.x * rinv; a2.y += v2.y * rinv; a2.z += v2.z * rinv; a2.w += v2.w * rinv;
    a3.x += v3.x * rinv; a3.y += v3.y * rinv; a3.z += v3.z * rinv; a3.w += v3.w * rinv;
  }
  const int d0 = lane * 4;
  atomicAdd(&lsum[0 * 128 + d0 + 0], a0.x); atomicAdd(&lsum[0 * 128 + d0 + 1], a0.y);
  atomicAdd(&lsum[0 * 128 + d0 + 2], a0.z); atomicAdd(&lsum[0 * 128 + d0 + 3], a0.w);
  atomicAdd(&lsum[1 * 128 + d0 + 0], a1.x); atomicAdd(&lsum[1 * 128 + d0 + 1], a1.y);
  atomicAdd(&lsum[1 * 128 + d0 + 2], a1.z); atomicAdd(&lsum[1 * 128 + d0 + 3], a1.w);
  atomicAdd(&lsum[2 * 128 + d0 + 0], a2.x); atomicAdd(&lsum[2 * 128 + d0 + 1], a2.y);
  atomicAdd(&lsum[2 * 128 + d0 + 2], a2.z); atomicAdd(&lsum[2 * 128 + d0 + 3], a2.w);
  atomicAdd(&lsum[3 * 128 + d0 + 0], a3.x); atomicAdd(&lsum[3 * 128 + d0 + 1], a3.y);
  atomicAdd(&lsum[3 * 128 + d0 + 2], a3.z); atomicAdd(&lsum[3 * 128 + d0 + 3], a3.w);
  __syncthreads();
  for (int i = tid; i < D_; i += 256) atomicAdd(&hsum[b * D_ + i], lsum[i]);
}

// ---------------- pass 2: M = Wq^T * Wk  (512x512x512) via fp32 WMMA ---------
// One wave per 16x16 output tile; K-loop of 128 V_WMMA_F32_16X16X4_F32.
__global__ void k_gram(const float* __restrict__ Wq, const float* __restrict__ Wk,
                       float* __restrict__ Mo) {
  const int gw = (blockIdx.x * blockDim.x + threadIdx.x) >> 5;  // 0..1023
  const int lane = threadIdx.x & 31;
  const int ib = (gw & 31) * 16, jb = (gw >> 5) * 16;
  const int half = lane >> 4, l = lane & 15;
#if __has_builtin(__builtin_amdgcn_wmma_f32_16x16x4_f32)
  v8f c = {0.f, 0.f, 0.f, 0.f, 0.f, 0.f, 0.f, 0.f};
  for (int kb = 0; kb < D_; kb += 4) {
    const int r0 = kb + half * 2;  // this half-wave holds K rows r0, r0+1
    v2f a, bm;
    a[0]  = Wq[(size_t)r0 * D_ + ib + l];
    a[1]  = Wq[(size_t)(r0 + 1) * D_ + ib + l];
    bm[0] = Wk[(size_t)r0 * D_ + jb + l];
    bm[1] = Wk[(size_t)(r0 + 1) * D_ + jb + l];
    c = __builtin_amdgcn_wmma_f32_16x16x4_f32(false, a, false, bm,
                                              (short)0, c, false, false);
  }
#pragma unroll
  for (int r = 0; r < 8; ++r)
    Mo[(size_t)(ib + half * 8 + r) * D_ + jb + l] = c[r];
#else
  float creg[8] = {0, 0, 0, 0, 0, 0, 0, 0};
  for (int k = 0; k < D_; ++k) {
    const float bv = Wk[(size_t)k * D_ + jb + l];
#pragma unroll
    for (int r = 0; r < 8; ++r)
      creg[r] += Wq[(size_t)k * D_ + ib + half * 8 + r] * bv;
  }
#pragma unroll
  for (int r = 0; r < 8; ++r)
    Mo[(size_t)(ib + half * 8 + r) * D_ + jb + l] = creg[r];
#endif
}

// ---------------- pass 3: u[b] = SCALE * M * hsum[b] -------------------------
__global__ void k_u(const float* __restrict__ Mo, const float* __restrict__ hsum,
                    float* __restrict__ u) {
  const int b = blockIdx.y;
  const int i = blockIdx.x * 256 + threadIdx.x;
  float acc = 0.f;
  const float* mr = Mo + (size_t)i * D_;
  const float* hs = hsum + b * D_;
  for (int j = 0; j < D_; ++j) acc += mr[j] * hs[j];
  u[b * D_ + i] = acc * SCALE_;
}

// ---------------- pass 4: scores -> hard boundary flags ----------------------
// sigmoid((score+logistic)/1) > 0.5  <=>  score + logistic > 0
__global__ void k_scores(const float* __restrict__ hidden,
                         const float* __restrict__ invnorm,
                         const float* __restrict__ noise,
                         const float* __restrict__ u,
                         int* __restrict__ flags) {
  __shared__ float us[D_];
  const int tid = threadIdx.x, lane = tid & 31, wave = tid >> 5;
  const int b = blockIdx.y;
  for (int i = tid; i < D_; i += 256) us[i] = u[b * D_ + i];
  __syncthreads();
  const int s0 = blockIdx.x * 64 + wave * 8;
  for (int t = 0; t < 8; ++t) {
    const int s = s0 + t;
    const float* hp = hidden + ((size_t)b * S_ + s) * D_;
    const float4* hp4 = (const float4*)hp;
    const float4* up4 = (const float4*)us;
    float d = 0.f;
#pragma unroll
    for (int j = 0; j < 4; ++j) {
      float4 v = hp4[j * 32 + lane];
      float4 w = up4[j * 32 + lane];
      d += v.x * w.x + v.y * w.y + v.z * w.z + v.w * w.w;
    }
    if (t + 1 < 8) __builtin_prefetch(hp + D_ + lane * 16, 0, 1);
    d = wave_sum(d);
    if (lane == 0) {
      const float score = d * invnorm[b * S_ + s];
      const float nz = noise[b * S_ + s];
      const float logistic = logf(nz) - log1pf(-nz);
      flags[b * S_ + s] = (score + logistic > 0.f) ? 1 : 0;
    }
  }
}

// ---------------- pass 5: per-batch stable compaction scan -------------------
// one block (1024 thr) per batch; each thread owns 8 contiguous tokens.
__global__ void k_scan(const int* __restrict__ flags, int* __restrict__ kept,
                       int* __restrict__ counts) {
  __shared__ int sd[1024];
  const int b = blockIdx.x, tid = threadIdx.x;
  const int base = b * S_ + tid * 8;
  int f[8], sum = 0;
#pragma unroll
  for (int e = 0; e < 8; ++e) { f[e] = flags[base + e]; sum += f[e]; }
  sd[tid] = sum;
  __syncthreads();
  for (int off = 1; off < 1024; off <<= 1) {
    int v = (tid >= off) ? sd[tid - off] : 0;
    __syncthreads();
    sd[tid] += v;
    __syncthreads();
  }
  if (tid == 1023) counts[b] = sd[1023];
  int run = sd[tid] - sum;  // exclusive prefix
#pragma unroll
  for (int e = 0; e < 8; ++e)
    if (f[e]) kept[b * S_ + run++] = tid * 8 + e;
}

// ---------------- pass 6: pooled output (gather kept rows, zero tail) --------
// grid (S/8, B), block 256 = 8 waves; one wave per output row.
__global__ void k_pool(const float* __restrict__ hidden,
                       const float* __restrict__ invnorm,
                       const int* __restrict__ kept,
                       const int* __restrict__ counts,
                       float* __restrict__ out) {
  const int tid = threadIdx.x, lane = tid & 31, wave = tid >> 5;
  const int b = blockIdx.y;
  const int i = blockIdx.x * 8 + wave;
  float4* op = (float4*)(out + ((size_t)b * S_ + i) * D_);
  if (i < counts[b]) {
    const int s = kept[b * S_ + i];
    const float rinv = invnorm[b * S_ + s];
    const float4* hp4 = (const float4*)(hidden + ((size_t)b * S_ + s) * D_);
#pragma unroll
    for (int j = 0; j < 4; ++j) {
      float4 v = hp4[j * 32 + lane];
      v.x *= rinv; v.y *= rinv; v.z *= rinv; v.w *= rinv;
      op[j * 32 + lane] = v;
    }
  } else {
    const float4 z = {0.f, 0.f, 0.f, 0.f};
#pragma unroll
    for (int j = 0; j < 4; ++j) op[j * 32 + lane] = z;
  }
}

// ---------------- pass 7: Binomial(S, PRIOR) log-prob loss -------------------
__global__ void k_loss(const int* __restrict__ counts, float* __restrict__ out_loss) {
  __shared__ float sh[B_];
  const int tid = threadIdx.x;
  if (tid < B_) {
    const float n = (float)S_;
    const float k = (float)counts[tid];
    sh[tid] = lgammaf(n + 1.f) - lgammaf(k + 1.f) - lgammaf(n - k + 1.f)
            + k * logf(PRIOR_) + (n - k) * log1pf(-PRIOR_);
  }
  __syncthreads();
  if (tid == 0) {
    float s = 0.f;
    for (int i = 0; i < B_; ++i) s += sh[i];
    *out_loss = -(s / (float)B_) / (float)S_;
  }
}

extern "C" void kernel_launch(void* const* d_in, const int* in_sizes, int n_in,
                              void* d_out, int out_size, void* d_ws, size_t ws_size,
                              hipStream_t stream) {
  (void)in_sizes; (void)n_in; (void)out_size; (void)ws_size;
  const float* hidden = (const float*)d_in[0];
  const float* Wq     = (const float*)d_in[1];
  const float* Wk     = (const float*)d_in[2];
  const float* noise  = (const float*)d_in[3];
  float* out = (float*)d_out;

  float* invnorm = (float*)d_ws;             // B*S
  float* hsum    = invnorm + B_ * S_;        // B*D
  float* Mo      = hsum + B_ * D_;           // D*D
  float* ub      = Mo + (size_t)D_ * D_;     // B*D
  int*   flags   = (int*)(ub + B_ * D_);     // B*S
  int*   kept    = flags + B_ * S_;          // B*S
  int*   counts  = kept + B_ * S_;           // B

  k_zero<<<(B_ * D_ + 255) / 256, 256, 0, stream>>>(hsum);
  k_norm_hsum<<<dim3(S_ / 64, B_), 256, 0, stream>>>(hidden, invnorm, hsum);
  k_gram<<<128, 256, 0, stream>>>(Wq, Wk, Mo);                 // 1024 waves = 1024 tiles
  k_u<<<dim3(D_ / 256, B_), 256, 0, stream>>>(Mo, hsum, ub);
  k_scores<<<dim3(S_ / 64, B_), 256, 0, stream>>>(hidden, invnorm, noise, ub, flags);
  k_scan<<<B_, 1024, 0, stream>>>(flags, kept, counts);
  k_pool<<<dim3(S_ / 8, B_), 256, 0, stream>>>(hidden, invnorm, kept, counts, out);
  k_loss<<<1, 32, 0, stream>>>(counts, out + (size_t)B_ * S_ * D_);
}